// Harris3DCurvature_8572754722951
// MI455X (gfx1250) — compile-verified
//
#include <hip/hip_runtime.h>
#include <hip/hip_bf16.h>

// ---------------- problem constants ----------------
constexpr int C = 16, D = 64, H = 96, W = 96;
constexpr int CH = D * H * W;          // 589824 floats per channel
constexpr int CH4 = CH / 4;            // 147456 float4 per channel
constexpr float K_HARRIS = 0.04f;
constexpr int KSEL = 8;                // RATIO * C

// tile config for the fused Harris pass
constexpr int DT = 4, HT = 12, WT = 32;
constexpr int TD = D / DT, TH = H / HT, TW = W / WT;   // 16, 8, 3
constexpr int NTILE = TD * TH * TW;                    // 384 tiles per channel
// LDS staging shapes
constexpr int XS_D = DT + 4, XS_H = HT + 4, XS_W = WT + 4;   // 8 x 16 x 36
constexpr int G_D = DT + 2, G_H = HT + 2, G_W = WT + 2;      // 6 x 14 x 34
constexpr int NXS = XS_D * XS_H * XS_W;                      // 4608
constexpr int NG  = G_D * G_H * G_W;                         // 2856
constexpr int NOUT = DT * HT * WT;                           // 1536

typedef float v2f __attribute__((ext_vector_type(2)));
typedef float v8f __attribute__((ext_vector_type(8)));
typedef unsigned int u32x4 __attribute__((ext_vector_type(4)));
typedef int i32x4 __attribute__((ext_vector_type(4)));
typedef int i32x8 __attribute__((ext_vector_type(8)));

#if defined(__has_builtin)
#if __has_builtin(__builtin_amdgcn_tensor_load_to_lds) && __has_builtin(__builtin_amdgcn_s_wait_tensorcnt)
#define USE_TDM 1
#endif
#endif

// Low 32 bits of a generic pointer to __shared__ = LDS byte offset
// (ISA 10.2: LDS aperture keeps the offset in ADDR[31:0]).
__device__ __forceinline__ unsigned lds_addr_u32(const void* p) {
    return (unsigned)(unsigned long long)p;
}

// ---------------------------------------------------------------------------
// Kernel 1: fused Sobel -> structure tensor -> 3x3x3 box -> Harris -> tile sum
// grid = (NTILE, C), block = 256
// ---------------------------------------------------------------------------
__global__ __launch_bounds__(256)
void harris_partial_kernel(const float* __restrict__ x, float* __restrict__ partial) {
    __shared__ float xs[NXS];
    __shared__ float gxs[NG];
    __shared__ float gys[NG];
    __shared__ float gzs[NG];
    __shared__ float red[256];

    const int tile = blockIdx.x;
    const int c    = blockIdx.y;
    const int tw = tile % TW;
    const int th = (tile / TW) % TH;
    const int td = tile / (TW * TH);
    const int d0 = td * DT, h0 = th * HT, w0 = tw * WT;
    const int tid = threadIdx.x;

    const float* xc = x + (size_t)c * CH;

    // ---- stage 1: stage x tile (halo 2) into LDS ---------------------------
    // Tiles whose low corner is in-bounds use the Tensor Data Mover: the
    // high-side halo is produced by TDM's tensor_dim OOB clip (reads beyond
    // the tensor return zero == the reference's zero padding). Low-border
    // tiles fall back to the manual zero-filling loop.
#ifdef USE_TDM
    const bool tdm_ok = (d0 >= 2) && (h0 >= 2) && (w0 >= 2);
    if (tdm_ok) {
        if (tid < 32) {   // one wave issues the DMA (TDM ignores EXEC)
            const float* gsrc = xc + ((size_t)(d0 - 2) * H + (h0 - 2)) * W + (w0 - 2);
            unsigned long long ga = (unsigned long long)gsrc;
            int rem0 = W - (w0 - 2);            // remaining extent along W
            int rem1 = H - (h0 - 2);            // remaining extent along H
            int rem2 = D - (d0 - 2);            // remaining extent along D

            u32x4 g0;
            g0[0] = 1u;                                  // count=1, user mode
            g0[1] = lds_addr_u32(xs);                    // LDS byte address
            g0[2] = (unsigned)(ga & 0xffffffffull);      // global_addr[31:0]
            g0[3] = (unsigned)((ga >> 32) & 0x1ffffffull) | (2u << 30); // addr[56:32] | type=2

            i32x8 g1;
            g1[0] = 2 << 16;                             // data_size = 4 bytes
            g1[1] = (rem0 & 0xffff) << 16;               // tensor_dim0[15:0]
            g1[2] = ((rem0 >> 16) & 0xffff) | ((rem1 & 0xffff) << 16); // dim0 hi | dim1 lo
            g1[3] = ((rem1 >> 16) & 0xffff) | (XS_W << 16);            // dim1 hi | tile_dim0
            g1[4] = XS_H | (XS_D << 16);                 // tile_dim1 | tile_dim2
            g1[5] = W;                                   // tensor_dim0_stride[31:0]
            g1[6] = ((H * W) & 0xffff) << 16;            // d0s hi(0) | tensor_dim1_stride lo
            g1[7] = (H * W) >> 16;                       // tensor_dim1_stride hi

            i32x4 g2;
            g2[0] = rem2;                                // tensor_dim2
            g2[1] = 0; g2[2] = 0; g2[3] = 0;             // tensor_dim3 / tile_dim3 unused

            i32x4 g3; g3[0] = 0; g3[1] = 0; g3[2] = 0; g3[3] = 0;

#if __clang_major__ >= 23
            i32x8 gz; gz[0]=0; gz[1]=0; gz[2]=0; gz[3]=0; gz[4]=0; gz[5]=0; gz[6]=0; gz[7]=0;
            __builtin_amdgcn_tensor_load_to_lds(g0, g1, g2, g3, gz, 0);
#else
            __builtin_amdgcn_tensor_load_to_lds(g0, g1, g2, g3, 0);
#endif
            __builtin_amdgcn_s_wait_tensorcnt(0);
        }
    } else
#endif
    {
        for (int i = tid; i < NXS; i += 256) {
            int lw = i % XS_W;
            int t  = i / XS_W;
            int lh = t % XS_H;
            int ld = t / XS_H;
            int gd = d0 - 2 + ld, gh = h0 - 2 + lh, gw = w0 - 2 + lw;
            float v = 0.0f;
            if ((unsigned)gd < (unsigned)D && (unsigned)gh < (unsigned)H && (unsigned)gw < (unsigned)W)
                v = xc[(gd * H + gh) * W + gw];
            xs[i] = v;
        }
    }
    __syncthreads();

    // ---- stage 2: Sobel gradients on halo-1 region -------------------------
    // grads outside the global domain are 0 (reference box-filters products
    // with zero padding); inside, computed from zero-padded x.
    for (int i = tid; i < NG; i += 256) {
        int lw = i % G_W;
        int t  = i / G_W;
        int lh = t % G_H;
        int ld = t / G_H;
        int gd = d0 - 1 + ld, gh = h0 - 1 + lh, gw = w0 - 1 + lw;
        float gxv = 0.f, gyv = 0.f, gzv = 0.f;
        if ((unsigned)gd < (unsigned)D && (unsigned)gh < (unsigned)H && (unsigned)gw < (unsigned)W) {
            float cs[3][3];
            float dz = 0.f;
            #pragma unroll
            for (int dy = 0; dy < 3; ++dy) {
                #pragma unroll
                for (int dx = 0; dx < 3; ++dx) {
                    int b = (ld * XS_H + (lh + dy)) * XS_W + (lw + dx);
                    float a0 = xs[b];
                    float a1 = xs[b + XS_H * XS_W];
                    float a2 = xs[b + 2 * XS_H * XS_W];
                    cs[dy][dx] = a0 + a1 + a2;
                    dz += a2 - a0;
                }
            }
            gzv = dz;
            gxv = (cs[0][2] - cs[0][0]) + 2.0f * (cs[1][2] - cs[1][0]) + (cs[2][2] - cs[2][0]);
            gyv = (cs[2][0] - cs[0][0]) + 2.0f * (cs[2][1] - cs[0][1]) + (cs[2][2] - cs[0][2]);
        }
        gxs[i] = gxv; gys[i] = gyv; gzs[i] = gzv;
    }
    __syncthreads();

    // ---- stage 3: box-filtered structure tensor + Harris, accumulate -------
    const float inv27 = 1.0f / 27.0f;
    float hsum = 0.0f;
    for (int i = tid; i < NOUT; i += 256) {
        int ow = i % WT;
        int t  = i / WT;
        int oh = t % HT;
        int od = t / HT;
        float sxx = 0.f, syy = 0.f, szz = 0.f, sxy = 0.f, sxz = 0.f, syz = 0.f;
        #pragma unroll
        for (int dz = 0; dz < 3; ++dz) {
            #pragma unroll
            for (int dy = 0; dy < 3; ++dy) {
                #pragma unroll
                for (int dx = 0; dx < 3; ++dx) {
                    int gi = ((od + dz) * G_H + (oh + dy)) * G_W + (ow + dx);
                    float ax = gxs[gi], ay = gys[gi], az = gzs[gi];
                    sxx += ax * ax; syy += ay * ay; szz += az * az;
                    sxy += ax * ay; sxz += ax * az; syz += ay * az;
                }
            }
        }
        float bxx = sxx * inv27, byy = syy * inv27, bzz = szz * inv27;
        float bxy = sxy * inv27, bxz = sxz * inv27, byz = syz * inv27;
        float det = bxx * (byy * bzz - byz * byz)
                  - bxy * (bxy * bzz - byz * bxz)
                  + bxz * (bxy * byz - byy * bxz);
        float tr = bxx + byy + bzz;
        hsum += det - K_HARRIS * tr * tr;
    }

    // ---- block reduction (deterministic tree) ------------------------------
    red[tid] = hsum;
    __syncthreads();
    for (int s = 128; s > 0; s >>= 1) {
        if (tid < s) red[tid] += red[tid + s];
        __syncthreads();
    }
    if (tid == 0) partial[tile * C + c] = red[0];
}

// ---------------------------------------------------------------------------
// Kernel 2: reduce partial[NTILE][16] over tiles with V_WMMA_F32_16X16X4_F32
// (channels = M rows, tiles = K, B = ones), then top-8 channel selection.
// launch: 1 block x 32 threads (one wave32, EXEC all ones for WMMA)
// ---------------------------------------------------------------------------
__global__ void reduce_topk_kernel(const float* __restrict__ partial,
                                   float* __restrict__ p_out,
                                   int* __restrict__ idx_out) {
    const int lane = threadIdx.x;
    const int m    = lane & 15;
    const int half = lane >> 4;

    v8f acc = {0.f, 0.f, 0.f, 0.f, 0.f, 0.f, 0.f, 0.f};
    v2f bones; bones.x = 1.0f; bones.y = 1.0f;

    // A(16x4): lanes 0-15 rows, vgpr0=K0,vgpr1=K1; lanes 16-31 vgpr0=K2,vgpr1=K3
    for (int t = 0; t < NTILE / 4; ++t) {
        int k0 = 4 * t + 2 * half;
        v2f a;
        a.x = partial[k0 * C + m];
        a.y = partial[(k0 + 1) * C + m];
        // D = A x ones + D ; column 0 of D accumulates p[channel]
        acc = __builtin_amdgcn_wmma_f32_16x16x4_f32(
            false, a, false, bones, (short)0, acc, false, false);
    }

    __shared__ float ps[C];
    // D layout: vgpr r, lanes 0-15 -> M=r; lanes 16-31 -> M=8+r. N = lane & 15.
    if (lane == 0) {
        #pragma unroll
        for (int r = 0; r < 8; ++r) ps[r] = acc[r];
    }
    if (lane == 16) {
        #pragma unroll
        for (int r = 0; r < 8; ++r) ps[8 + r] = acc[r];
    }
    __syncthreads();

    if (lane == 0) {
        bool used[C];
        #pragma unroll
        for (int i = 0; i < C; ++i) used[i] = false;
        for (int j = 0; j < KSEL; ++j) {
            int best = -1;
            float bv = 0.0f;
            for (int cc = 0; cc < C; ++cc) {
                if (!used[cc] && (best < 0 || ps[cc] > bv)) { bv = ps[cc]; best = cc; }
            }
            used[best] = true;
            idx_out[j] = best;
            p_out[j] = bv;
        }
    }
}

// ---------------------------------------------------------------------------
// Kernel 3: gather the 8 selected channels (pure bandwidth, float4 copies)
// ---------------------------------------------------------------------------
__global__ __launch_bounds__(256)
void gather_kernel(const float4* __restrict__ x4,
                   const int* __restrict__ idx,
                   float4* __restrict__ out4) {
    int t = blockIdx.x * 256 + threadIdx.x;      // [0, 8*CH4)
    int j = t / CH4;
    int r = t - j * CH4;
    int ci = idx[j];
    const float4* src = x4 + (size_t)ci * CH4 + r;
    __builtin_prefetch(src + 4096, 0, 0);        // global_prefetch_b8, stream ahead
    out4[t] = *src;
}

// ---------------------------------------------------------------------------
extern "C" void kernel_launch(void* const* d_in, const int* in_sizes, int n_in,
                              void* d_out, int out_size, void* d_ws, size_t ws_size,
                              hipStream_t stream) {
    const float* x = (const float*)d_in[0];   // [1,16,64,96,96] fp32
    float* out = (float*)d_out;               // [1,8,64,96,96] fp32

    float* wsf     = (float*)d_ws;
    float* partial = wsf;                       // NTILE*C floats
    float* p_vals  = wsf + NTILE * C;           // 16 floats (top-8 values)
    int*   idx     = (int*)(wsf + NTILE * C + 16);

    dim3 grid1(NTILE, C);
    harris_partial_kernel<<<grid1, 256, 0, stream>>>(x, partial);

    reduce_topk_kernel<<<1, 32, 0, stream>>>(partial, p_vals, idx);

    int total4 = KSEL * CH4;                    // 1,179,648 float4
    gather_kernel<<<total4 / 256, 256, 0, stream>>>((const float4*)x, idx, (float4*)out);
    (void)in_sizes; (void)n_in; (void)out_size; (void)ws_size;
}